// mixprop_27212912787591
// MI455X (gfx1250) — compile-verified
//
#include <hip/hip_runtime.h>
#include <hip/hip_bf16.h>

typedef __attribute__((ext_vector_type(2))) float v2f;
typedef __attribute__((ext_vector_type(8))) float v8f;
typedef __attribute__((ext_vector_type(4))) unsigned int v4u;
typedef __attribute__((ext_vector_type(8))) int v8i;
typedef __attribute__((ext_vector_type(4))) int v4i;

#define B_  8
#define CIN 32
#define N_  1024
#define C_  16
#define HID 20
#define GDEP 3
#define COUT 32
#define BC (B_ * C_)          // 128
#define ROWS (BC * N_)        // 131072
#define HP 32                 // global h row stride (cols 20..31 are zero)
#define LP 40                 // LDS h row stride (TDM pad: +8 dwords per 32)
#define ALPHA 0.05f
#define BETA  0.95f

#ifndef __has_builtin
#define __has_builtin(x) 0
#endif

// ---------------------------------------------------------------------------
// Kernel 1: h0[b,c,n,0:20] = sum_i x[b,i,n,c] * Ws[i,:] + bs ; cols 20..31 = 0
// ---------------------------------------------------------------------------
__global__ __launch_bounds__(256) void h0_kernel(
    const float* __restrict__ x, const float* __restrict__ Ws,
    const float* __restrict__ bs, float* __restrict__ h0) {
  int tid = blockIdx.x * blockDim.x + threadIdx.x;
  if (tid >= ROWS) return;
  int c = tid & 15;
  int n = (tid >> 4) & (N_ - 1);
  int b = tid >> 14;
  float acc[HID];
#pragma unroll
  for (int j = 0; j < HID; ++j) acc[j] = bs[j];
  for (int i = 0; i < CIN; ++i) {
    float xv = x[(((size_t)b * CIN + i) * N_ + n) * C_ + c];
#pragma unroll
    for (int j = 0; j < HID; ++j) acc[j] += xv * Ws[i * HID + j];
  }
  float* dst = h0 + (((size_t)b * C_ + c) * N_ + n) * HP;
#pragma unroll
  for (int j = 0; j < HID; ++j) dst[j] = acc[j];
#pragma unroll
  for (int j = HID; j < HP; ++j) dst[j] = 0.f;
}

// ---------------------------------------------------------------------------
// Kernel 2: inv[row] = 1 / (1 + sum_w adj[row, w])   (one wave32 per row)
// adj is streamed once with zero reuse -> non-temporal loads.
// ---------------------------------------------------------------------------
__global__ __launch_bounds__(256) void inv_kernel(
    const float* __restrict__ adj, float* __restrict__ inv) {
  int gtid = blockIdx.x * blockDim.x + threadIdx.x;
  int row  = gtid >> 5;
  int lane = gtid & 31;
  if (row >= ROWS) return;
  const float* r = adj + (size_t)row * N_;
  float s = 0.f;
#pragma unroll 8
  for (int i = lane; i < N_; i += 32) s += __builtin_nontemporal_load(r + i);
#pragma unroll
  for (int off = 16; off > 0; off >>= 1) s += __shfl_xor(s, off, 32);
  if (lane == 0) inv[row] = 1.0f / (1.0f + s);
}

// ---------------------------------------------------------------------------
// Kernel 3: one propagation step.
//   hout[v,:] = ALPHA*h0[v,:] + BETA*inv[v]*( adj[v,:]·hin + hin[v,:] )
// grid = (8 rowblocks, 128 bc); block = 256 (8 waves x 16 rows).
// hin panel (1024 x 32 f32) is DMA'd into LDS by the Tensor Data Mover,
// one 4096-dword chunk per wave, with TDM padding (+8 dwords per 32) so LDS
// rows have stride 40 floats -> bank-conflict-free B-fragment reads.
// WMMA f32 16x16x4; 2 accumulators cover cols 0..15 / 16..31 (zeros beyond 20).
// ---------------------------------------------------------------------------
__global__ __launch_bounds__(256) void prop_kernel(
    const float* __restrict__ adj, const float* __restrict__ inv,
    const float* __restrict__ h0, const float* __restrict__ hin,
    float* __restrict__ hout) {
  __shared__ float panel[N_ * LP];  // 163840 bytes

  int bc = blockIdx.y;
  int rowblk = blockIdx.x;
  int wave = threadIdx.x >> 5;
  int lane = threadIdx.x & 31;

#if __has_builtin(__builtin_amdgcn_tensor_load_to_lds)
  {
    // Per-wave TDM chunk: 4096 dwords = 128 source rows of 32 floats.
    unsigned lds_base =
        (unsigned)(uintptr_t)panel + (unsigned)wave * (128u * LP * 4u);
    unsigned long long ga = (unsigned long long)(uintptr_t)(
        hin + (size_t)bc * N_ * HP + (size_t)wave * 4096);
    v4u g0;
    g0.x = 1u;                                   // count=1, user descriptor
    g0.y = lds_base;                             // lds_addr (bytes)
    g0.z = (unsigned)(ga & 0xffffffffu);         // global_addr[31:0]
    g0.w = (unsigned)((ga >> 32) & 0x1ffffffu)   // global_addr[56:32]
           | (2u << 30);                         // type = 2 ("image")
    v8i g1;
    g1[0] = (int)((2u << 16)      // data_size = 4B
                  | (1u << 20)    // pad_enable
                  | (4u << 22)    // pad_interval: every 32 dwords
                  | (7u << 25));  // pad_amount: 8 dwords  -> row stride 40
    g1[1] = (int)((4096u & 0xffffu) << 16);  // tensor_dim0 = 4096 (lo16)
    g1[2] = (int)(1u << 16);                 // dim0 hi=0 | tensor_dim1 = 1
    g1[3] = (int)(4096u << 16);              // dim1 hi=0 | tile_dim0 = 4096
    g1[4] = 0;                               // tile_dim1 = 0, tile_dim2 = 0
    g1[5] = 4096;                            // tensor_dim0_stride lo32
    g1[6] = 0;
    g1[7] = 0;
    v4i gz4 = {};
    v8i gz8 = {};
    __builtin_amdgcn_tensor_load_to_lds(g0, g1, gz4, gz4, gz8, 0);
    __builtin_amdgcn_s_wait_tensorcnt(0);
  }
#else
  {
    const float4* src = (const float4*)(hin + (size_t)bc * N_ * HP);
    for (int i = threadIdx.x; i < N_ * HP / 4; i += 256) {
      float4 vv = src[i];
      int w = (i * 4) / HP, col = (i * 4) % HP;
      *(float4*)&panel[w * LP + col] = vv;
    }
  }
#endif
  __syncthreads();

  int m = lane & 15;
  int sel2 = (lane >> 4) << 1;  // 0 for lanes 0-15, 2 for lanes 16-31
  int rbase = rowblk * 128 + wave * 16;

  const float* arow = adj + ((size_t)bc * N_ + rbase + m) * N_;

  v8f acc0 = {};
  v8f acc1 = {};
  for (int kb = 0; kb < N_; kb += 4) {
    int kr = kb + sel2;
    v2f a = __builtin_nontemporal_load((const v2f*)(arow + kr));
    const float* p = &panel[kr * LP + m];
    v2f b0, b1;
    b0.x = p[0];
    b0.y = p[LP];
    b1.x = p[16];
    b1.y = p[LP + 16];
    acc0 = __builtin_amdgcn_wmma_f32_16x16x4_f32(false, a, false, b0,
                                                 (short)0, acc0, false, false);
    acc1 = __builtin_amdgcn_wmma_f32_16x16x4_f32(false, a, false, b1,
                                                 (short)0, acc1, false, false);
  }

  // Epilogue. C/D layout: VGPR r -> row rbase+r (lanes 0-15) / rbase+r+8
  // (lanes 16-31), column = lane&15 (+16 for tile 1).
  // For j>=20: h0==0, panel==0, acc1==0 -> stores 0, preserving padding.
  int rowoff = (lane >= 16) ? 8 : 0;
#pragma unroll
  for (int r = 0; r < 8; ++r) {
    int v = rbase + r + rowoff;
    size_t gbase = ((size_t)bc * N_ + v) * HP;
    float iv = inv[(size_t)bc * N_ + v];
    float val0 = ALPHA * h0[gbase + m] + BETA * iv * (acc0[r] + panel[v * LP + m]);
    hout[gbase + m] = val0;
    int j = 16 + m;
    float val1 = ALPHA * h0[gbase + j] + BETA * iv * (acc1[r] + panel[v * LP + j]);
    hout[gbase + j] = val1;
  }
}

// ---------------------------------------------------------------------------
// Kernel 4: out[row, :] = concat(h0..h3)[row, 0:20 each] (80) · W_end + b_end
// One wave per 16 rows; W_end cached in LDS; 20 WMMA k-steps (k=4) which
// never straddle a 20-wide source slice (20 % 4 == 0).
// ---------------------------------------------------------------------------
__global__ __launch_bounds__(256) void out_kernel(
    const float* __restrict__ h0, const float* __restrict__ h1,
    const float* __restrict__ h2, const float* __restrict__ h3,
    const float* __restrict__ We, const float* __restrict__ be,
    float* __restrict__ out) {
  __shared__ float w[HID * (GDEP + 1) * COUT];  // 80*32 = 10 KB
  __shared__ float bsh[COUT];
  for (int i = threadIdx.x; i < 80 * COUT; i += 256) w[i] = We[i];
  if (threadIdx.x < COUT) bsh[threadIdx.x] = be[threadIdx.x];
  __syncthreads();

  int wave = threadIdx.x >> 5;
  int lane = threadIdx.x & 31;
  int m = lane & 15;
  int sel2 = (lane >> 4) << 1;
  size_t rbase = ((size_t)blockIdx.x * 8 + wave) * 16;

  const float* hb[4] = {h0, h1, h2, h3};
  v8f acc0 = {};
  v8f acc1 = {};
#pragma unroll
  for (int kb = 0; kb < 80; kb += 4) {
    const float* hs = hb[kb / HID];
    int ko = kb % HID;
    v2f a = *(const v2f*)(hs + (rbase + m) * HP + ko + sel2);
    int kr = kb + sel2;
    v2f b0, b1;
    b0.x = w[kr * COUT + m];
    b0.y = w[(kr + 1) * COUT + m];
    b1.x = w[kr * COUT + 16 + m];
    b1.y = w[(kr + 1) * COUT + 16 + m];
    acc0 = __builtin_amdgcn_wmma_f32_16x16x4_f32(false, a, false, b0,
                                                 (short)0, acc0, false, false);
    acc1 = __builtin_amdgcn_wmma_f32_16x16x4_f32(false, a, false, b1,
                                                 (short)0, acc1, false, false);
  }

  int rowoff = (lane >= 16) ? 8 : 0;
#pragma unroll
  for (int r = 0; r < 8; ++r) {
    size_t v = rbase + r + rowoff;
    out[v * COUT + m] = acc0[r] + bsh[m];
    out[v * COUT + 16 + m] = acc1[r] + bsh[16 + m];
  }
}

// ---------------------------------------------------------------------------
extern "C" void kernel_launch(void* const* d_in, const int* in_sizes, int n_in,
                              void* d_out, int out_size, void* d_ws,
                              size_t ws_size, hipStream_t stream) {
  const float* x   = (const float*)d_in[0];  // [8,32,1024,16]
  const float* adj = (const float*)d_in[1];  // [8,16,1024,1024]
  const float* Ws  = (const float*)d_in[2];  // [32,20]
  const float* bs  = (const float*)d_in[3];  // [20]
  const float* We  = (const float*)d_in[4];  // [80,32]
  const float* be  = (const float*)d_in[5];  // [32]
  float* out = (float*)d_out;                // [8,16,1024,32]

  const size_t HN = (size_t)ROWS * HP;  // 4,194,304 floats per h buffer
  float* inv = (float*)d_ws;            // 131072 floats
  float* h0  = inv + ROWS;
  float* h1  = h0 + HN;
  float* h2  = h1 + HN;
  float* h3  = h2 + HN;

  h0_kernel<<<(ROWS + 255) / 256, 256, 0, stream>>>(x, Ws, bs, h0);
  inv_kernel<<<(ROWS * 32) / 256, 256, 0, stream>>>(adj, inv);

  dim3 g(8, BC);  // 8 row-blocks of 128 rows, 128 (b,c) pairs
  prop_kernel<<<g, 256, 0, stream>>>(adj, inv, h0, h0, h1);
  prop_kernel<<<g, 256, 0, stream>>>(adj, inv, h0, h1, h2);
  prop_kernel<<<g, 256, 0, stream>>>(adj, inv, h0, h2, h3);

  out_kernel<<<ROWS / 128, 256, 0, stream>>>(h0, h1, h2, h3, We, be, out);
}